// GridEncoderMinkowskiHierarchical_74706661147205
// MI455X (gfx1250) — compile-verified
//
#include <hip/hip_runtime.h>
#include <hip/hip_bf16.h>
#include <stdint.h>

// instant-NGP style hierarchical hash-grid encoder for gfx1250 (MI455X).
//
// Layout: one lane per (query, level). 256-thread blocks = 16 queries x 16
// levels. Wave32 => one wave covers 2 queries x 16 levels, so the final
// float2 store is a fully contiguous 256B burst per wave.
//
// CDNA5-specific paths used:
//   * global_load_async_to_lds_b32 (ASYNCcnt) to stage the block's query
//     coordinates straight into LDS without a VGPR round trip.
//   * s_wait_asynccnt 0 split-counter wait.
//   * non-temporal output stores (th:TH_STORE_NT) so the 128MB write-once
//     output does not evict the 64MB embedding table from the 192MB L2.

#define NUM_LEVELS 16
#define LEVEL_DIM 2
#define LOG2_HASHMAP 19
#define TABLE_SIZE (1u << LOG2_HASHMAP)
#define TABLE_MASK (TABLE_SIZE - 1u)
#define BLOCK_THREADS 256
#define QUERIES_PER_BLOCK (BLOCK_THREADS / NUM_LEVELS)  // 16

#define PRIME1 2654435761u
#define PRIME2 805459861u

typedef float v2f __attribute__((ext_vector_type(2)));

__global__ __launch_bounds__(BLOCK_THREADS) void
hashgrid_encode_gfx1250(const float* __restrict__ inputs,     // [N,3]
                        const v2f* __restrict__ embeddings,   // [L, T] float2
                        const int* __restrict__ bound_ptr,    // scalar
                        float* __restrict__ out,              // [N, L*2]
                        int n_queries) {
  __shared__ float s_coords[QUERIES_PER_BLOCK * 3];

  const int tid = threadIdx.x;
  const int q0 = blockIdx.x * QUERIES_PER_BLOCK;

  // ---- async stage of this block's coordinates into LDS (CDNA5 path) ----
  const int q_in_block =
      (n_queries - q0 >= QUERIES_PER_BLOCK) ? QUERIES_PER_BLOCK : (n_queries - q0);
  const int nfloats = q_in_block * 3;  // <= 48
  if (tid < nfloats) {
    const float* gsrc = inputs + (size_t)q0 * 3 + tid;
    uint32_t lds_addr = (uint32_t)(uintptr_t)(&s_coords[tid]);
    asm volatile("global_load_async_to_lds_b32 %0, %1, off"
                 :
                 : "v"(lds_addr), "v"(gsrc)
                 : "memory");
  }
  asm volatile("s_wait_asynccnt 0" ::: "memory");
  __syncthreads();

  const int qi = tid >> 4;     // query slot within block
  const int level = tid & 15;  // hash level
  const int q = q0 + qi;
  if (q >= n_queries) return;

  const float inv_bound = 1.0f / (float)bound_ptr[0];

  // reference permutes channels: x = inputs[..., [2,0,1]]
  const float x0 = s_coords[qi * 3 + 2];
  const float x1 = s_coords[qi * 3 + 0];
  const float x2 = s_coords[qi * 3 + 1];

  // PER_LEVEL_SCALE == 2.0 exactly -> res = 16 << level (exact).
  const float resf = (float)(16 << level);

  // pos = ((x/bound + 1) * 0.5) * res - 0.5  (align_corners=False)
  float p, fl;
  p = (x0 * inv_bound + 1.0f) * 0.5f * resf - 0.5f;
  fl = floorf(p);
  const float fr0 = p - fl;
  const uint32_t c0 = (uint32_t)(int)fl;

  p = (x1 * inv_bound + 1.0f) * 0.5f * resf - 0.5f;
  fl = floorf(p);
  const float fr1 = p - fl;
  const uint32_t c1 = (uint32_t)(int)fl;

  p = (x2 * inv_bound + 1.0f) * 0.5f * resf - 0.5f;
  fl = floorf(p);
  const float fr2 = p - fl;
  const uint32_t c2 = (uint32_t)(int)fl;

  // hash components per corner offset (PRIME0 == 1)
  const uint32_t h0a = c0, h0b = c0 + 1u;
  const uint32_t h1a = c1 * PRIME1, h1b = (c1 + 1u) * PRIME1;
  const uint32_t h2a = c2 * PRIME2, h2b = (c2 + 1u) * PRIME2;

  const float w0a = 1.0f - fr0;
  const float w1a = 1.0f - fr1;
  const float w2a = 1.0f - fr2;

  const v2f* __restrict__ tbl = embeddings + (size_t)level * TABLE_SIZE;

  // 8 independent b64 gathers, fully unrolled so all 8 are in flight before
  // any accumulation waits (gather-latency hiding is the whole game here).
  float r0 = 0.0f, r1 = 0.0f;
#pragma unroll
  for (int c = 0; c < 8; ++c) {
    const uint32_t h = ((c & 4) ? h0b : h0a) ^ ((c & 2) ? h1b : h1a) ^
                       ((c & 1) ? h2b : h2a);
    const v2f f = tbl[h & TABLE_MASK];
    const float w = ((c & 4) ? fr0 : w0a) * ((c & 2) ? fr1 : w1a) *
                    ((c & 1) ? fr2 : w2a);
    r0 = fmaf(w, f.x, r0);
    r1 = fmaf(w, f.y, r1);
  }

  // [N, L*2] level-major concat; contiguous 256B burst per wave32.
  v2f o;
  o.x = r0;
  o.y = r1;
  v2f* dst = (v2f*)(out + (size_t)q * (NUM_LEVELS * LEVEL_DIM) + level * LEVEL_DIM);
  __builtin_nontemporal_store(o, dst);
}

extern "C" void kernel_launch(void* const* d_in, const int* in_sizes, int n_in,
                              void* d_out, int out_size, void* d_ws,
                              size_t ws_size, hipStream_t stream) {
  const float* inputs = (const float*)d_in[0];
  const v2f* embeddings = (const v2f*)d_in[1];
  const int* bound_ptr = (const int*)d_in[2];
  float* out = (float*)d_out;

  const int n_queries = in_sizes[0] / 3;
  const int blocks = (n_queries + QUERIES_PER_BLOCK - 1) / QUERIES_PER_BLOCK;

  hashgrid_encode_gfx1250<<<blocks, BLOCK_THREADS, 0, stream>>>(
      inputs, embeddings, bound_ptr, out, n_queries);
}